// DynamicNMS_36507222016519
// MI455X (gfx1250) — compile-verified
//
#include <hip/hip_runtime.h>

#define NBOX   4096
#define BLOCK  1024
#define MAXDET 300
#define NW     (NBOX / 32)

// LDS layout (dynamic shared):
//   s_box  : NBOX float4  (raw boxes -> offset boxes -> sorted offset boxes)   65536 B
//   s_key  : NBOX floats  (sort keys)                                          16384 B
//   s_idx  : NBOX ints    (argsort payload)                                    16384 B
//   s_area : NBOX floats  (areas of sorted boxes)                              16384 B
//   s_keep : NW   uints   (keep bitmask over sorted positions)                   512 B
//   s_wp   : NW   ints    (per-word exclusive rank prefix)                       512 B
#define SMEM_BYTES (NBOX * 28 + NW * 8)

__global__ __launch_bounds__(BLOCK) void DynamicNMS_kernel(
    const float* __restrict__ boxes,      // (B, N, 4)
    const float* __restrict__ scores,     // (B, N)
    const int*   __restrict__ labels,     // (B, N)
    const float* __restrict__ nms_thresh, // scalar
    const float* __restrict__ cw,         // (2,)
    const int*   __restrict__ img_size,   // scalar
    float* __restrict__ out, int B)
{
    extern __shared__ char smem[];
    float4*   s_box  = (float4*)(smem);                  // NBOX
    float*    s_key  = (float*)(smem + NBOX * 16);       // NBOX
    int*      s_idx  = (int*)  (smem + NBOX * 20);       // NBOX
    float*    s_area = (float*)(smem + NBOX * 24);       // NBOX
    unsigned* s_keep = (unsigned*)(smem + NBOX * 28);    // NW
    int*      s_wp   = (int*)  (smem + NBOX * 28 + NW * 4);

    const int b   = blockIdx.x;
    const int tid = threadIdx.x;

    const float fimg   = (float)(*img_size);
    const float loff_c = fimg + 1.0f;
    const float iou_th = 1.0f / (1.0f + __expf(-nms_thresh[0]));
    const float w0 = cw[0], w1 = cw[1];
    const float NEG_INF = -__builtin_inff();

    const float* gbox = boxes  + (size_t)b * NBOX * 4;
    const float* gsc  = scores + (size_t)b * NBOX;
    const int*   glb  = labels + (size_t)b * NBOX;

    // ---------------- Phase 0: async stage raw boxes (64KB) into LDS --------
    // gfx1250 ASYNCcnt path: memory -> LDS without a VGPR round-trip.
    {
        unsigned ldsBase = (unsigned)(unsigned long long)(uintptr_t)s_box;
        unsigned long long gb64 = (unsigned long long)(uintptr_t)gbox;
        #pragma unroll
        for (int k = 0; k < (NBOX * 16) / (16 * BLOCK); ++k) {
            unsigned goff = (unsigned)(tid + k * BLOCK) * 16u;
            unsigned loff = ldsBase + goff;
            asm volatile("global_load_async_to_lds_b128 %0, %1, %2"
                         :: "v"(loff), "v"(goff), "s"(gb64)
                         : "memory");
        }
        asm volatile("s_wait_asynccnt 0" ::: "memory");
    }
    __syncthreads();

    // ---------------- Phase 1: clip, class-offset, sort keys ----------------
    for (int i = tid; i < NBOX; i += BLOCK) {
        float4 bx = s_box[i];                         // ds_load_b128
        int   l = glb[i];
        float s = gsc[i];
        float lo = (float)l * loff_c;                 // separates classes for NMS
        float4 ob;
        ob.x = fminf(fmaxf(bx.x, 0.0f), fimg) + lo;
        ob.y = fminf(fmaxf(bx.y, 0.0f), fimg) + lo;
        ob.z = fminf(fmaxf(bx.z, 0.0f), fimg) + lo;
        ob.w = fminf(fmaxf(bx.w, 0.0f), fimg) + lo;
        s_box[i] = ob;                                // ds_store_b128
        s_key[i] = (s > 0.3f) ? s * ((l == 0) ? w0 : w1) : NEG_INF;
        s_idx[i] = i;
    }
    __syncthreads();

    // ---------------- Phase 2: bitonic argsort (descending, idx tiebreak) ---
    for (int k = 2; k <= NBOX; k <<= 1) {
        for (int j = k >> 1; j > 0; j >>= 1) {
            for (int i = tid; i < NBOX; i += BLOCK) {
                int p = i ^ j;
                if (p > i) {
                    bool up = ((i & k) == 0);
                    float ka = s_key[i], kb = s_key[p];
                    int   ia = s_idx[i], ib = s_idx[p];
                    bool beforeAB = (ka > kb) || (ka == kb && ia < ib);
                    if (up ? !beforeAB : beforeAB) {
                        s_key[i] = kb; s_key[p] = ka;
                        s_idx[i] = ib; s_idx[p] = ia;
                    }
                }
            }
            __syncthreads();
        }
    }

    // ---------------- Phase 3: gather boxes into sorted order ---------------
    float4 gx[NBOX / BLOCK];
    #pragma unroll
    for (int k = 0; k < NBOX / BLOCK; ++k) {
        int p = tid + k * BLOCK;
        gx[k] = s_box[s_idx[p]];                      // ds_load_b128 (gather)
    }
    __syncthreads();
    #pragma unroll
    for (int k = 0; k < NBOX / BLOCK; ++k) {
        int p = tid + k * BLOCK;
        s_box[p]  = gx[k];                            // ds_store_b128
        s_area[p] = fmaxf(gx[k].z - gx[k].x, 0.0f) *
                    fmaxf(gx[k].w - gx[k].y, 0.0f);
    }
    // keep bitmask init: valid == finite key (sorted order)
    for (int w = tid; w < NW; w += BLOCK) {
        unsigned m = 0u;
        #pragma unroll 8
        for (int t = 0; t < 32; ++t)
            if (s_key[w * 32 + t] != NEG_INF) m |= (1u << t);
        s_keep[w] = m;
    }
    __syncthreads();

    // ---------------- Phase 4: greedy NMS ----------------------------------
    for (int i = 0; i < NBOX; ++i) {
        // Uniform scalar read: all clears happened before the last barrier.
        if (!((s_keep[i >> 5] >> (i & 31)) & 1u)) continue;   // no writes -> no barrier
        float4 bi = s_box[i];
        float  ai = s_area[i];
        for (int p = i + 1 + tid; p < NBOX; p += BLOCK) {
            if ((s_keep[p >> 5] >> (p & 31)) & 1u) {
                float4 bp = s_box[p];                 // ds_load_b128
                float iw = fminf(bi.z, bp.z) - fmaxf(bi.x, bp.x);
                float ih = fminf(bi.w, bp.w) - fmaxf(bi.y, bp.y);
                float inter = fmaxf(iw, 0.0f) * fmaxf(ih, 0.0f);
                float denom = ai + s_area[p] - inter + 1e-9f;  // > 0
                // inter/denom > th  <=>  inter > th*denom  (division-free)
                if (inter > iou_th * denom)
                    atomicAnd(&s_keep[p >> 5], ~(1u << (p & 31)));
            }
        }
        __syncthreads();
    }
    __syncthreads();

    // ---------------- Phase 5: rank (popcount prefix) -----------------------
    for (int w = tid; w < NW; w += BLOCK) s_wp[w] = __popc(s_keep[w]);
    __syncthreads();
    if (tid == 0) {
        int acc = 0;
        for (int w = 0; w < NW; ++w) { int t = s_wp[w]; s_wp[w] = acc; acc += t; }
    }
    __syncthreads();

    // ---------------- Phase 6: zero + scatter outputs ------------------------
    float*  oboxes  = out;
    float*  oscores = out + (size_t)B * MAXDET * 4;
    float*  olabels = out + (size_t)B * MAXDET * 5;
    float*  ovalid  = out + (size_t)B * MAXDET * 6;
    float4* obox4   = (float4*)oboxes;

    for (int q = tid; q < MAXDET; q += BLOCK) {
        obox4[b * MAXDET + q]   = make_float4(0.0f, 0.0f, 0.0f, 0.0f);
        oscores[b * MAXDET + q] = 0.0f;
        olabels[b * MAXDET + q] = 0.0f;
        ovalid [b * MAXDET + q] = 0.0f;
    }
    __syncthreads();

    for (int p = tid; p < NBOX; p += BLOCK) {
        unsigned word = s_keep[p >> 5];
        if ((word >> (p & 31)) & 1u) {
            int rank = s_wp[p >> 5] + __popc(word & ((1u << (p & 31)) - 1u));
            if (rank < MAXDET) {
                int j = s_idx[p];
                float4 rb = ((const float4*)gbox)[j];       // global_load_b128
                float4 cb;
                cb.x = fminf(fmaxf(rb.x, 0.0f), fimg);
                cb.y = fminf(fmaxf(rb.y, 0.0f), fimg);
                cb.z = fminf(fmaxf(rb.z, 0.0f), fimg);
                cb.w = fminf(fmaxf(rb.w, 0.0f), fimg);
                obox4[b * MAXDET + rank] = cb;              // global_store_b128
                oscores[b * MAXDET + rank] = gsc[j];
                olabels[b * MAXDET + rank] = (float)glb[j];
                ovalid [b * MAXDET + rank] = 1.0f;
            }
        }
    }
}

extern "C" void kernel_launch(void* const* d_in, const int* in_sizes, int n_in,
                              void* d_out, int out_size, void* d_ws, size_t ws_size,
                              hipStream_t stream) {
    const float* boxes  = (const float*)d_in[0];
    const float* scores = (const float*)d_in[1];
    const int*   labels = (const int*)  d_in[2];
    const float* nth    = (const float*)d_in[3];
    const float* cw     = (const float*)d_in[4];
    const int*   img    = (const int*)  d_in[5];

    int B = in_sizes[1] / NBOX;   // scores is (B, N)
    if (B < 1) B = 1;

    DynamicNMS_kernel<<<B, BLOCK, SMEM_BYTES, stream>>>(
        boxes, scores, labels, nth, cw, img, (float*)d_out, B);
}